// HandSynthesizerCano_71519795413461
// MI455X (gfx1250) — compile-verified
//
#include <hip/hip_runtime.h>
#include <math.h>

// ---------------------------------------------------------------------------
// HandSynthesizerCano forward for MI455X (gfx1250, wave32, WMMA).
// Shapes: B=4 T=16 (BT=64) J=20 K=300 N_CROP=2048 N_FULL=16384
//         D_OBJ=256 D_EMB=64 D_JF=64, radius=0.025
// Point-MLP+maxpool stages use V_WMMA_F32_16X16X4_F32 (fp32 matrix pipe,
// K=6 padded to 8 via two chained WMMAs). bias+ReLU hoisted out of the max
// loops by monotonicity: max_p relu(x_p+b) == relu(b + max_p x_p).
// Ball query uses wave32 ballot scan (== argsort-of-unique-keys semantics).
// ---------------------------------------------------------------------------

typedef __attribute__((ext_vector_type(2))) float v2f;
typedef __attribute__((ext_vector_type(8))) float v8f;

#define BTN    64
#define JJ     20
#define KK     300
#define KPAD   304          // 19 * 16
#define NCROP  2048
#define NFULL  16384
#define DOBJ   256
#define DEMB   64
#define DJF    64
#define RAD2   (0.025f * 0.025f)
#define NEGINF (-INFINITY)

static __device__ __forceinline__ unsigned ballot32(bool p) {
  return __builtin_amdgcn_ballot_w32(p);
}

// D = A(16x4,f32) * B(4x16,f32) + C(16x16,f32)
static __device__ __forceinline__ v8f wmma_k4(v2f a, v2f b, v8f c) {
  return __builtin_amdgcn_wmma_f32_16x16x4_f32(
      /*neg_a=*/false, a, /*neg_b=*/false, b,
      /*c_mod=*/(short)0, c, /*reuse_a=*/false, /*reuse_b=*/false);
}

// ---------------------------------------------------------------------------
// K1: obj_feat[bt,n] = max_{p<2048} relu(obj_crop[bt,p,:6] @ W_obj[:,n] + b_obj[n])
// grid = 64 bt * 8 tile-pairs, block = 32 (one wave per 32 output columns:
// each A tile feeds 4 WMMAs -> better WMMA/VMEM ratio)
// ---------------------------------------------------------------------------
__global__ void k_objfeat(const float* __restrict__ obj_crop,
                          const float* __restrict__ W_obj,
                          const float* __restrict__ b_obj,
                          float* __restrict__ obj_feat) {
  const int bt   = blockIdx.x >> 3;
  const int np   = blockIdx.x & 7;          // pair of 16-col tiles
  const int lane = threadIdx.x;
  const int c0   = np * 32 + (lane & 15);
  const int c1   = c0 + 16;
  const bool lo  = lane < 16;

  // B tiles: lanes 0-15 hold K rows {0,1}, lanes 16-31 rows {2,3}; tile1 rows 6,7 = 0
  const int r0 = lo ? 0 : 2, r1 = lo ? 1 : 3;
  v2f b0a, b0b, b1a, b1b;
  b0a.x = W_obj[r0*DOBJ + c0]; b0a.y = W_obj[r1*DOBJ + c0];
  b0b.x = W_obj[r0*DOBJ + c1]; b0b.y = W_obj[r1*DOBJ + c1];
  b1a.x = 0.f; b1a.y = 0.f; b1b.x = 0.f; b1b.y = 0.f;
  if (lo) {
    b1a.x = W_obj[4*DOBJ + c0]; b1a.y = W_obj[5*DOBJ + c0];
    b1b.x = W_obj[4*DOBJ + c1]; b1b.y = W_obj[5*DOBJ + c1];
  }

  const float* base = obj_crop + (size_t)bt * NCROP * 6;
  v8f mxa = {NEGINF,NEGINF,NEGINF,NEGINF,NEGINF,NEGINF,NEGINF,NEGINF};
  v8f mxb = mxa;

  for (int mt = 0; mt < NCROP / 16; ++mt) {
    const float* p = base + (size_t)(mt * 16 + (lane & 15)) * 6;
    __builtin_prefetch(p + 16 * 6, 0, 1);   // next point tile -> global_prefetch
    // A tile: lanes 0-15 channels {0,1}/{4,5}; lanes 16-31 channels {2,3}/{0,0}
    v2f a0 = *(const v2f*)(p + (lo ? 0 : 2));
    v2f a1; a1.x = 0.f; a1.y = 0.f;
    if (lo) a1 = *(const v2f*)(p + 4);

    v8f c = {0.f,0.f,0.f,0.f,0.f,0.f,0.f,0.f};
    c = wmma_k4(a0, b0a, c);
    c = wmma_k4(a1, b1a, c);
    v8f d = {0.f,0.f,0.f,0.f,0.f,0.f,0.f,0.f};
    d = wmma_k4(a0, b0b, d);
    d = wmma_k4(a1, b1b, d);
#pragma unroll
    for (int r = 0; r < 8; ++r) {           // raw running max; bias+relu deferred
      mxa[r] = fmaxf(mxa[r], c[r]);
      mxb[r] = fmaxf(mxb[r], d[r]);
    }
  }
  float ma = mxa[0], mb = mxb[0];
#pragma unroll
  for (int r = 1; r < 8; ++r) { ma = fmaxf(ma, mxa[r]); mb = fmaxf(mb, mxb[r]); }
  ma = fmaxf(ma, __shfl_xor(ma, 16, 32));   // combine rows r and r+8 halves
  mb = fmaxf(mb, __shfl_xor(mb, 16, 32));
  if (lo) {
    obj_feat[bt * DOBJ + c0] = fmaxf(ma + b_obj[c0], 0.f);
    obj_feat[bt * DOBJ + c1] = fmaxf(mb + b_obj[c1], 0.f);
  }
}

// ---------------------------------------------------------------------------
// K2: init MLP -> joints, Rodrigues IK (R,t), analytic rigid inverse, joint embed
// grid = 64 (bt), block = 64
// ---------------------------------------------------------------------------
__global__ void k_ik(const float* __restrict__ traj,
                     const float* __restrict__ obj_feat,
                     const float* __restrict__ tmpl,
                     const float* __restrict__ W_init, const float* __restrict__ b_init,
                     const float* __restrict__ W_emb,  const float* __restrict__ b_emb,
                     float* __restrict__ jcano, float* __restrict__ Rm,
                     float* __restrict__ Rinv,  float* __restrict__ tinv,
                     float* __restrict__ emb,   float* __restrict__ out_joints) {
  const int bt  = blockIdx.x;
  const int tid = threadIdx.x;
  __shared__ float sfeat[265];
  __shared__ float sj[60];

  for (int i = tid; i < 265; i += 64)
    sfeat[i] = (i < 9) ? traj[bt * 9 + i] : obj_feat[bt * DOBJ + (i - 9)];
  __syncthreads();

  if (tid < 60) {
    float acc = b_init[tid];
    for (int i = 0; i < 265; ++i) acc = fmaf(sfeat[i], W_init[i * 60 + tid], acc);
    sj[tid] = acc;
    jcano[bt * 60 + tid] = acc;
    out_joints[bt * 60 + tid] = acc;   // joints_init_out
  }
  __syncthreads();

  if (tid >= 1 && tid <= 20) {
    const int j = tid;
    const float eps = 1e-8f;
    float t0x = tmpl[0], t0y = tmpl[1], t0z = tmpl[2];
    float tjx = tmpl[j*3+0], tjy = tmpl[j*3+1], tjz = tmpl[j*3+2];
    float jx = sj[(j-1)*3+0], jy = sj[(j-1)*3+1], jz = sj[(j-1)*3+2];

    float ax = tjx - t0x, ay = tjy - t0y, az = tjz - t0z;
    float bx = jx, by = jy, bz = jz;      // joints21 root is exactly zero
    float an = sqrtf(ax*ax + ay*ay + az*az) + eps; ax /= an; ay /= an; az /= an;
    float bn = sqrtf(bx*bx + by*by + bz*bz) + eps; bx /= bn; by /= bn; bz /= bn;

    float vx = ay*bz - az*by, vy = az*bx - ax*bz, vz = ax*by - ay*bx;
    float cth = ax*bx + ay*by + az*bz;
    float s2  = vx*vx + vy*vy + vz*vz;
    float kf  = (1.f - cth) / (s2 + eps);

    float M[9] = {0.f,-vz, vy,  vz, 0.f,-vx,  -vy, vx, 0.f};
    float M2[9];
#pragma unroll
    for (int r = 0; r < 3; ++r)
#pragma unroll
      for (int c = 0; c < 3; ++c)
        M2[r*3+c] = M[r*3+0]*M[0*3+c] + M[r*3+1]*M[1*3+c] + M[r*3+2]*M[2*3+c];

    float R[9];
#pragma unroll
    for (int r = 0; r < 3; ++r)
#pragma unroll
      for (int c = 0; c < 3; ++c)
        R[r*3+c] = (r == c ? 1.f : 0.f) + M[r*3+c] + M2[r*3+c] * kf;
    if (s2 < eps) {
#pragma unroll
      for (int i = 0; i < 9; ++i) R[i] = 0.f;
      R[0] = R[4] = R[8] = 1.f;
    }

    float tx = jx - (R[0]*tjx + R[1]*tjy + R[2]*tjz);
    float ty = jy - (R[3]*tjx + R[4]*tjy + R[5]*tjz);
    float tz = jz - (R[6]*tjx + R[7]*tjy + R[8]*tjz);

    // 3x3 inverse via adjugate (G^-1 = [R^-1, -R^-1 t; 0 1])
    float c00 = R[4]*R[8] - R[5]*R[7];
    float c01 = R[5]*R[6] - R[3]*R[8];
    float c02 = R[3]*R[7] - R[4]*R[6];
    float det = R[0]*c00 + R[1]*c01 + R[2]*c02;
    float id  = 1.f / det;
    float Ri[9];
    Ri[0] = c00 * id;
    Ri[1] = (R[2]*R[7] - R[1]*R[8]) * id;
    Ri[2] = (R[1]*R[5] - R[2]*R[4]) * id;
    Ri[3] = c01 * id;
    Ri[4] = (R[0]*R[8] - R[2]*R[6]) * id;
    Ri[5] = (R[2]*R[3] - R[0]*R[5]) * id;
    Ri[6] = c02 * id;
    Ri[7] = (R[1]*R[6] - R[0]*R[7]) * id;
    Ri[8] = (R[0]*R[4] - R[1]*R[3]) * id;
    float tix = -(Ri[0]*tx + Ri[1]*ty + Ri[2]*tz);
    float tiy = -(Ri[3]*tx + Ri[4]*ty + Ri[5]*tz);
    float tiz = -(Ri[6]*tx + Ri[7]*ty + Ri[8]*tz);

    const int bj = bt * JJ + (j - 1);
#pragma unroll
    for (int i = 0; i < 9; ++i) { Rm[bj*9+i] = R[i]; Rinv[bj*9+i] = Ri[i]; }
    tinv[bj*3+0] = tix; tinv[bj*3+1] = tiy; tinv[bj*3+2] = tiz;
  }
  __syncthreads();

  for (int idx = tid; idx < JJ * DEMB; idx += 64) {
    const int j = idx >> 6, e = idx & 63;
    float x = sj[j*3], y = sj[j*3+1], z = sj[j*3+2];
    emb[(bt*JJ + j)*DEMB + e] =
        b_emb[e] + x*W_emb[e] + y*W_emb[64+e] + z*W_emb[128+e];
  }
}

// ---------------------------------------------------------------------------
// K3: ball query. First KK ascending in-radius indices; pad to KPAD with -1.
// grid = 1280 (bt*20), block = 256 (8 waves)
// ---------------------------------------------------------------------------
__global__ void k_ballquery(const float* __restrict__ obj_full,
                            const float* __restrict__ jcano,
                            int* __restrict__ ind) {
  const int bj = blockIdx.x;
  const int bt = bj / JJ;
  const int tid = threadIdx.x;
  const int lane = tid & 31, w = tid >> 5;
  __shared__ int s_cnt[8], s_off[8];
  __shared__ int s_base;
  if (tid == 0) s_base = 0;
  const float qx = jcano[bj*3+0], qy = jcano[bj*3+1], qz = jcano[bj*3+2];
  const float* pb = obj_full + (size_t)bt * NFULL * 6;
  int* op = ind + bj * KPAD;
  __syncthreads();

  for (int chunk = 0; chunk < NFULL / 256; ++chunk) {
    if (s_base >= KK) break;              // uniform: read after barrier
    const int i = chunk * 256 + tid;
    const float* p = pb + (size_t)i * 6;
    float dx = p[0]-qx, dy = p[1]-qy, dz = p[2]-qz;
    bool within = (dx*dx + dy*dy + dz*dz) < RAD2;
    unsigned msk = ballot32(within);
    int lp = __popc(msk & ((1u << lane) - 1u));
    if (lane == 0) s_cnt[w] = __popc(msk);
    __syncthreads();
    if (tid == 0) {
      int off = s_base;
#pragma unroll
      for (int ww = 0; ww < 8; ++ww) { s_off[ww] = off; off += s_cnt[ww]; }
      s_base = off;
    }
    __syncthreads();
    if (within) {
      int pos = s_off[w] + lp;
      if (pos < KK) op[pos] = i;
    }
  }
  __syncthreads();
  int total = s_base; if (total > KK) total = KK;
  for (int pos = tid; pos < KPAD; pos += 256)
    if (pos >= total) op[pos] = -1;
}

// ---------------------------------------------------------------------------
// K4: f[bj,n] = max_{k valid} relu( [Ginv*(p,1); Ginv_rot*n] @ W_jf[:,n] + b_jf[n] )
// grid = 5120 (bt*20*4 ntiles), block = 32 (one wave). Invalid rows -> -inf
// before max; bias+relu applied once at the end (relu(-inf+b) == 0).
// ---------------------------------------------------------------------------
__global__ void k_jointfeat(const float* __restrict__ obj_full,
                            const int* __restrict__ ind,
                            const float* __restrict__ Rinv,
                            const float* __restrict__ tinv,
                            const float* __restrict__ W_jf,
                            const float* __restrict__ b_jf,
                            float* __restrict__ fout) {
  const int nt  = blockIdx.x & 3;
  const int bj  = blockIdx.x >> 2;
  const int bt  = bj / JJ;
  const int lane = threadIdx.x;
  const int col  = nt * 16 + (lane & 15);
  const bool lo  = lane < 16;

  v2f b0, b1;
  if (lo) { b0.x = W_jf[0*DJF+col]; b0.y = W_jf[1*DJF+col];
            b1.x = W_jf[4*DJF+col]; b1.y = W_jf[5*DJF+col]; }
  else    { b0.x = W_jf[2*DJF+col]; b0.y = W_jf[3*DJF+col];
            b1.x = 0.f;             b1.y = 0.f; }

  float Ri[9];
#pragma unroll
  for (int i = 0; i < 9; ++i) Ri[i] = Rinv[bj*9 + i];
  const float tx = tinv[bj*3+0], ty = tinv[bj*3+1], tz = tinv[bj*3+2];

  const int*   indp = ind + bj * KPAD;
  const float* ofb  = obj_full + (size_t)bt * NFULL * 6;
  const int half = lo ? 0 : 8;
  v8f mx = {NEGINF,NEGINF,NEGINF,NEGINF,NEGINF,NEGINF,NEGINF,NEGINF};

  for (int mt = 0; mt < KPAD / 16; ++mt) {
    const int s  = mt * 16 + (lane & 15);
    const int iv = indp[s];
    const bool valid = iv >= 0;
    const int g  = valid ? iv : (NFULL - 1);
    const float* p = ofb + (size_t)g * 6;
    v2f p01 = *(const v2f*)(p + 0);
    v2f p23 = *(const v2f*)(p + 2);
    v2f p45 = *(const v2f*)(p + 4);
    float x = p01.x, y = p01.y, z = p23.x;
    float nx = p23.y, ny = p45.x, nz = p45.y;
    // canonicalize: cp = Rinv*p + tinv ; cn = Rinv*n
    float cpx = Ri[0]*x + Ri[1]*y + Ri[2]*z + tx;
    float cpy = Ri[3]*x + Ri[4]*y + Ri[5]*z + ty;
    float cpz = Ri[6]*x + Ri[7]*y + Ri[8]*z + tz;
    float cnx = Ri[0]*nx + Ri[1]*ny + Ri[2]*nz;
    float cny = Ri[3]*nx + Ri[4]*ny + Ri[5]*nz;
    float cnz = Ri[6]*nx + Ri[7]*ny + Ri[8]*nz;
    // sp channels: [cpx cpy cpz cnx cny cnz], K padded to 8
    v2f a0, a1;
    if (lo) { a0.x = cpx; a0.y = cpy; a1.x = cny; a1.y = cnz; }
    else    { a0.x = cpz; a0.y = cnx; a1.x = 0.f; a1.y = 0.f; }
    v8f c = {0.f,0.f,0.f,0.f,0.f,0.f,0.f,0.f};
    c = wmma_k4(a0, b0, c);
    c = wmma_k4(a1, b1, c);
    const unsigned vsh = ballot32(valid) >> half;  // bit r = validity of row r+half
#pragma unroll
    for (int r = 0; r < 8; ++r) {
      float v = ((vsh >> r) & 1u) ? c[r] : NEGINF; // masked raw max
      mx[r] = fmaxf(mx[r], v);
    }
  }
  float m = mx[0];
#pragma unroll
  for (int r = 1; r < 8; ++r) m = fmaxf(m, mx[r]);
  m = fmaxf(m, __shfl_xor(m, 16, 32));
  if (lo) fout[bj * DJF + col] = fmaxf(m + b_jf[col], 0.f);  // deferred bias+relu
}

// ---------------------------------------------------------------------------
// K5: disp_cano = [emb ; f](128) @ W_st(128,3); disp = R * disp_cano
// (the "- joints_init + jcano" terms cancel exactly: identical arrays)
// grid = 1280 (bt*20), block = 32
// ---------------------------------------------------------------------------
__global__ void k_final(const float* __restrict__ emb,
                        const float* __restrict__ f,
                        const float* __restrict__ W_st,
                        const float* __restrict__ Rm,
                        float* __restrict__ out_disp) {
  const int bj = blockIdx.x;
  const int lane = threadIdx.x;
  float a0 = 0.f, a1 = 0.f, a2 = 0.f;
#pragma unroll
  for (int s = 0; s < 4; ++s) {
    const int d = lane + 32 * s;
    float fd = (d < 64) ? emb[bj*64 + d] : f[bj*64 + (d - 64)];
    a0 = fmaf(fd, W_st[d*3+0], a0);
    a1 = fmaf(fd, W_st[d*3+1], a1);
    a2 = fmaf(fd, W_st[d*3+2], a2);
  }
#pragma unroll
  for (int off = 16; off > 0; off >>= 1) {
    a0 += __shfl_xor(a0, off, 32);
    a1 += __shfl_xor(a1, off, 32);
    a2 += __shfl_xor(a2, off, 32);
  }
  if (lane == 0) {
    const float* R = Rm + bj * 9;
    out_disp[bj*3+0] = R[0]*a0 + R[1]*a1 + R[2]*a2;
    out_disp[bj*3+1] = R[3]*a0 + R[4]*a1 + R[5]*a2;
    out_disp[bj*3+2] = R[6]*a0 + R[7]*a1 + R[8]*a2;
  }
}

// ---------------------------------------------------------------------------
extern "C" void kernel_launch(void* const* d_in, const int* in_sizes, int n_in,
                              void* d_out, int out_size, void* d_ws, size_t ws_size,
                              hipStream_t stream) {
  const float* traj   = (const float*)d_in[0];
  const float* crop   = (const float*)d_in[1];
  const float* full   = (const float*)d_in[2];
  const float* tmpl   = (const float*)d_in[3];
  const float* W_obj  = (const float*)d_in[4];
  const float* b_obj  = (const float*)d_in[5];
  const float* W_init = (const float*)d_in[6];
  const float* b_init = (const float*)d_in[7];
  const float* W_emb  = (const float*)d_in[8];
  const float* b_emb  = (const float*)d_in[9];
  const float* W_jf   = (const float*)d_in[10];
  const float* b_jf   = (const float*)d_in[11];
  const float* W_st   = (const float*)d_in[12];
  float* out = (float*)d_out;

  // workspace layout (floats / ints), total ~2.4 MB
  float* ws       = (float*)d_ws;
  float* obj_feat = ws;                      // 64*256      = 16384
  float* jcano    = obj_feat + 16384;        // 64*20*3     = 3840
  float* Rm       = jcano    + 3840;         // 64*20*9     = 11520
  float* Rinv     = Rm       + 11520;        // 64*20*9     = 11520
  float* tinvp    = Rinv     + 11520;        // 64*20*3     = 3840
  float* emb      = tinvp    + 3840;         // 64*20*64    = 81920
  float* fout     = emb      + 81920;        // 64*20*64    = 81920
  int*   ind      = (int*)(fout + 81920);    // 64*20*304   = 389120 ints

  k_objfeat  <<<dim3(BTN * 8),       dim3(32),  0, stream>>>(crop, W_obj, b_obj, obj_feat);
  k_ik       <<<dim3(BTN),           dim3(64),  0, stream>>>(traj, obj_feat, tmpl, W_init, b_init,
                                                             W_emb, b_emb, jcano, Rm, Rinv, tinvp,
                                                             emb, out);
  k_ballquery<<<dim3(BTN * JJ),      dim3(256), 0, stream>>>(full, jcano, ind);
  k_jointfeat<<<dim3(BTN * JJ * 4),  dim3(32),  0, stream>>>(full, ind, Rinv, tinvp, W_jf, b_jf, fout);
  k_final    <<<dim3(BTN * JJ),      dim3(32),  0, stream>>>(emb, fout, W_st, Rm, out + BTN * JJ * 3);
}